// hideandseek_49057116455425
// MI455X (gfx1250) — compile-verified
//
#include <hip/hip_runtime.h>
#include <cstdint>

#define DROP_THRESH 3

// Native vector type so __builtin_nontemporal_load/store lower cleanly to
// global_load_b128 / global_store_b128 with th:TH_*_NT.
typedef float  vfloat4 __attribute__((ext_vector_type(4)));

__global__ __launch_bounds__(256) void hideandseek_49057116455425_kernel(
    const vfloat4* __restrict__ img4,
    const int*     __restrict__ dmask,   // [16*16]
    const int*     __restrict__ gate,    // [1]
    vfloat4*       __restrict__ out4,
    int n4)                              // element count / 4
{
    __shared__ int   mask_i[256];
    __shared__ float keep_f[256];

    const int t = threadIdx.x;

    // ---- Stage the 16x16 drop mask into LDS via CDNA5 async global->LDS ----
    // 256 threads x b32 = 1 KB, tracked with ASYNCcnt (CDNA5 ISA).
#if __has_builtin(__builtin_amdgcn_global_load_async_to_lds_b32)
    __builtin_amdgcn_global_load_async_to_lds_b32((int*)(dmask + t), &mask_i[t], 0, 0);
  #if __has_builtin(__builtin_amdgcn_s_wait_asynccnt)
    __builtin_amdgcn_s_wait_asynccnt(0);
  #else
    asm volatile("s_wait_asynccnt 0" ::: "memory");
  #endif
#else
    mask_i[t] = dmask[t];
#endif
    __syncthreads();

    // Fold gate into a per-cell float multiplier: 1.0 keeps, 0.0 drops.
    // gate==0 -> all cells 1.0 (value-exact pass-through).
    const int g = gate[0];
    keep_f[t] = (g == 0 || mask_i[t] >= DROP_THRESH) ? 1.0f : 0.0f;
    __syncthreads();

    // ---- Streaming body: one float4 per lane per iteration, NT hints ----
    const int stride = gridDim.x * blockDim.x;
    int j = blockIdx.x * blockDim.x + t;
    #pragma unroll 4
    for (; j < n4; j += stride) {
        // h = (j%128)*4 .. +3 all inside one 32-pixel cell (4 divides 32).
        const int hcell = (j >> 3)  & 15;          // (h/32)
        const int wcell = (j >> 12) & 15;          // (w/32)
        const float m = keep_f[(wcell << 4) | hcell];

        vfloat4 v = __builtin_nontemporal_load(&img4[j]);
        v *= m;
        __builtin_nontemporal_store(v, &out4[j]);
    }
}

extern "C" void kernel_launch(void* const* d_in, const int* in_sizes, int n_in,
                              void* d_out, int out_size, void* d_ws, size_t ws_size,
                              hipStream_t stream) {
    (void)n_in; (void)d_ws; (void)ws_size; (void)out_size;

    const float* img   = (const float*)d_in[0];   // [64,3,512,512] fp32
    const int*   dmask = (const int*)d_in[1];     // [16,16] int32
    const int*   gate  = (const int*)d_in[2];     // scalar int

    float* out = (float*)d_out;

    const int n  = in_sizes[0];    // 50,331,648 (divisible by 4)
    const int n4 = n >> 2;         // 12,582,912 float4s

    constexpr int TPB = 256;       // 8 wave32s per block
    constexpr int F4_PER_THREAD = 4;
    const int blocks = (n4 + TPB * F4_PER_THREAD - 1) / (TPB * F4_PER_THREAD);

    hideandseek_49057116455425_kernel<<<blocks, TPB, 0, stream>>>(
        (const vfloat4*)img, dmask, gate, (vfloat4*)out, n4);
}